// QattenNet_70385924047055
// MI455X (gfx1250) — compile-verified
//
#include <hip/hip_runtime.h>

typedef __attribute__((ext_vector_type(2))) float v2f;
typedef __attribute__((ext_vector_type(8))) float v8f;

#define NE 64
#define NT 128
#define NA 32
#define NH 4
#define NS 256
#define NO 128
#define NQH1 128
#define NQD 64
#define NC1 128
#define NB (NE * NT) // 8192

// ---------------------------------------------------------------------------
// Kernel 0: fold heads out of Wk / bk.
//   WkT[k][o] = sum_h Wk[h][o][k]   (QD x O, row-major: B-operand friendly)
//   bkSum[k]  = sum_h bk[h][k]
// ---------------------------------------------------------------------------
__global__ __launch_bounds__(256) void wk_reduce_kernel(
    const float* __restrict__ Wk, const float* __restrict__ bk,
    float* __restrict__ WkT, float* __restrict__ bkSum) {
  int idx = blockIdx.x * 256 + threadIdx.x; // over QD*O = 8192
  if (idx < NQD * NO) {
    int k = idx / NO;
    int o = idx % NO;
    float s = 0.f;
#pragma unroll
    for (int h = 0; h < NH; ++h)
      s += Wk[(size_t)h * NO * NQD + (size_t)o * NQD + k];
    WkT[idx] = s;
  }
  if (idx < NQD) {
    float s = 0.f;
#pragma unroll
    for (int h = 0; h < NH; ++h) s += bk[h * NQD + idx];
    bkSum[idx] = s;
  }
}

// ---------------------------------------------------------------------------
// WMMA fragment helpers (V_WMMA_F32_16X16X4_F32, wave32).
// A 16x4 layout: lane&15 = M, (lane>>4)*2 = K base, 2 VGPRs = K, K+1.
// B 4x16 layout: lane&15 = N, (lane>>4)*2 = K base, 2 VGPRs = K, K+1.
// C/D 16x16:     lane&15 = N, VGPR v -> M = v + 8*(lane>>4).
// ---------------------------------------------------------------------------
__device__ inline v2f load_a_lds(const float* At, int lda, int k0, int lane) {
  int m = lane & 15;
  int k = k0 + ((lane >> 4) << 1);
  const float* p = At + m * lda + k;
  v2f a;
  a.x = p[0];
  a.y = p[1];
  return a;
}

__device__ inline v2f load_b_glb(const float* Bm, int ldb, int k0, int n0, int lane) {
  int n = n0 + (lane & 15);
  int k = k0 + ((lane >> 4) << 1);
  v2f b;
  b.x = Bm[(size_t)k * ldb + n];
  b.y = Bm[(size_t)(k + 1) * ldb + n];
  return b;
}

template <int K>
__device__ inline v8f gemm_tile(const float* At, int lda,
                                const float* Bm, int ldb, int n0, int lane) {
  v8f acc = {};
#pragma unroll 4
  for (int k = 0; k < K; k += 4) {
    v2f a = load_a_lds(At, lda, k, lane);
    v2f b = load_b_glb(Bm, ldb, k, n0, lane);
    acc = __builtin_amdgcn_wmma_f32_16x16x4_f32(false, a, false, b, (short)0,
                                                acc, false, false);
  }
  return acc;
}

__device__ inline void store_tile_lds(float* Ht, int ldh, int n0, int lane,
                                      v8f acc, const float* __restrict__ bias,
                                      bool relu) {
  int n = n0 + (lane & 15);
  int mb = (lane >> 4) << 3;
  float bv = bias[n];
#pragma unroll
  for (int v = 0; v < 8; ++v) {
    float x = acc[v] + bv;
    Ht[(mb + v) * ldh + n] = relu ? fmaxf(x, 0.f) : x;
  }
}

// ---------------------------------------------------------------------------
// Stage 1: one wave per 16 rows of B. Chained WMMA GEMMs:
//   h1  = relu(s*Wq1+bq1)        16x256 * 256x128
//   emb = h1*Wq2+bq2             16x128 * 128x64
//   g   = emb*WkT                16x64  * 64x128   -> global ws
//   hc  = relu(s*Wc1+bc1)        16x256 * 256x128
//   cb  = hc*Wc2+bc2 ; sb = emb . bkSum            -> global ws
// ---------------------------------------------------------------------------
__global__ __launch_bounds__(32) void stage1_kernel(
    const float* __restrict__ states,
    const float* __restrict__ Wq1, const float* __restrict__ bq1,
    const float* __restrict__ Wq2, const float* __restrict__ bq2,
    const float* __restrict__ Wc1, const float* __restrict__ bc1,
    const float* __restrict__ Wc2, const float* __restrict__ bc2,
    const float* __restrict__ WkT, const float* __restrict__ bkSum,
    float* __restrict__ g, float* __restrict__ cb, float* __restrict__ sb) {
  constexpr int SP = NS + 4;   // 260: padded row stride (bank-conflict free)
  constexpr int HP = NQH1 + 4; // 132
  constexpr int EP = NQD + 4;  // 68
  __shared__ float sT[16 * SP];
  __shared__ float hT[16 * HP];
  __shared__ float eT[16 * EP];

  const int b0 = blockIdx.x * 16;
  const int lane = threadIdx.x;

  // Coalesced load of the 16x256 state tile into padded LDS.
  for (int i = lane; i < 16 * (NS / 4); i += 32) {
    int m = i / (NS / 4);
    int c4 = i % (NS / 4);
    float4 v = ((const float4*)(states + (size_t)(b0 + m) * NS))[c4];
    float* dst = &sT[m * SP + c4 * 4];
    dst[0] = v.x; dst[1] = v.y; dst[2] = v.z; dst[3] = v.w;
  }
  __syncthreads();

  // GEMM1: h1 = relu(s * Wq1 + bq1)
#pragma unroll
  for (int nt = 0; nt < NQH1 / 16; ++nt) {
    v8f acc = gemm_tile<NS>(sT, SP, Wq1, NQH1, nt * 16, lane);
    store_tile_lds(hT, HP, nt * 16, lane, acc, bq1, true);
  }
  __syncthreads();

  // GEMM2: emb = h1 * Wq2 + bq2
#pragma unroll
  for (int nt = 0; nt < NQD / 16; ++nt) {
    v8f acc = gemm_tile<NQH1>(hT, HP, Wq2, NQD, nt * 16, lane);
    store_tile_lds(eT, EP, nt * 16, lane, acc, bq2, false);
  }
  __syncthreads();

  // GEMM3: g = emb * WkT  -> straight to global workspace
#pragma unroll
  for (int nt = 0; nt < NO / 16; ++nt) {
    v8f acc = gemm_tile<NQD>(eT, EP, WkT, NO, nt * 16, lane);
    int n = nt * 16 + (lane & 15);
    int mb = (lane >> 4) << 3;
#pragma unroll
    for (int v = 0; v < 8; ++v)
      g[(size_t)(b0 + mb + v) * NO + n] = acc[v];
  }
  __syncthreads();

  // GEMM4: hc = relu(s * Wc1 + bc1)  (reuses hT)
#pragma unroll
  for (int nt = 0; nt < NC1 / 16; ++nt) {
    v8f acc = gemm_tile<NS>(sT, SP, Wc1, NC1, nt * 16, lane);
    store_tile_lds(hT, HP, nt * 16, lane, acc, bc1, true);
  }
  __syncthreads();

  // c = hc * Wc2 + bc2 ; sb = emb . bkSum  (N=1 -> scalar dots, 16 lanes)
  if (lane < 16) {
    float accc = 0.f;
    for (int k = 0; k < NC1; ++k) accc += hT[lane * HP + k] * Wc2[k];
    cb[b0 + lane] = accc + bc2[0];
    float sbv = 0.f;
    for (int k = 0; k < NQD; ++k) sbv += eT[lane * EP + k] * bkSum[k];
    sb[b0 + lane] = sbv;
  }
}

// ---------------------------------------------------------------------------
// Stage 2: stream f once (134 MB, the roofline cost). One block per b.
//   out[b] = sum_a q[b,a] * (f[b,a,:] . g[b,:]) + sb[b]*sum_a q[b,a] + cb[b]
// Thread t: agent a = t>>3, 16 contiguous o's = 4 float4 loads.
// ---------------------------------------------------------------------------
__global__ __launch_bounds__(256) void stage2_kernel(
    const float* __restrict__ q_values, const float* __restrict__ f,
    const float* __restrict__ g, const float* __restrict__ cb,
    const float* __restrict__ sb, float* __restrict__ out) {
  __shared__ float gS[NO];
  __shared__ float red[8];
  const int b = blockIdx.x;
  const int t = threadIdx.x;

  if (t < NO) gS[t] = g[(size_t)b * NO + t];
  __syncthreads();

  const int a = t >> 3;
  const int ob = (t & 7) * 16;
  const float4* fp = (const float4*)(f + ((size_t)b * NA + a) * NO + ob);
  float p = 0.f;
#pragma unroll
  for (int i = 0; i < 4; ++i) {
    float4 fv = fp[i];
    int o = ob + i * 4;
    p += fv.x * gS[o] + fv.y * gS[o + 1] + fv.z * gS[o + 2] + fv.w * gS[o + 3];
  }
  p *= q_values[(size_t)b * NA + a];

  // wave32 reduction, then cross-wave via LDS
  for (int off = 16; off > 0; off >>= 1) p += __shfl_down(p, off, 32);
  if ((t & 31) == 0) red[t >> 5] = p;
  __syncthreads();
  if (t == 0) {
    float s = 0.f;
#pragma unroll
    for (int w = 0; w < 8; ++w) s += red[w];
    float qsum = 0.f;
    for (int aa = 0; aa < NA; ++aa) qsum += q_values[(size_t)b * NA + aa];
    out[b] = s + sb[b] * qsum + cb[b];
  }
}

// ---------------------------------------------------------------------------
extern "C" void kernel_launch(void* const* d_in, const int* in_sizes, int n_in,
                              void* d_out, int out_size, void* d_ws,
                              size_t ws_size, hipStream_t stream) {
  const float* q_values = (const float*)d_in[0];
  const float* states = (const float*)d_in[1];
  const float* fs = (const float*)d_in[2];
  const float* Wq1 = (const float*)d_in[3];
  const float* bq1 = (const float*)d_in[4];
  const float* Wq2 = (const float*)d_in[5];
  const float* bq2 = (const float*)d_in[6];
  const float* Wk = (const float*)d_in[7];
  const float* bk = (const float*)d_in[8];
  const float* Wc1 = (const float*)d_in[9];
  const float* bc1 = (const float*)d_in[10];
  const float* Wc2 = (const float*)d_in[11];
  const float* bc2 = (const float*)d_in[12];
  float* out = (float*)d_out;

  // Workspace layout (~4.3 MB total)
  char* ws = (char*)d_ws;
  float* WkT = (float*)ws;   ws += (size_t)NQD * NO * sizeof(float);
  float* bkSum = (float*)ws; ws += (size_t)NQD * sizeof(float);
  float* g = (float*)ws;     ws += (size_t)NB * NO * sizeof(float);
  float* cb = (float*)ws;    ws += (size_t)NB * sizeof(float);
  float* sb = (float*)ws;

  wk_reduce_kernel<<<(NQD * NO + 255) / 256, 256, 0, stream>>>(Wk, bk, WkT,
                                                               bkSum);
  stage1_kernel<<<NB / 16, 32, 0, stream>>>(states, Wq1, bq1, Wq2, bq2, Wc1,
                                            bc1, Wc2, bc2, WkT, bkSum, g, cb,
                                            sb);
  stage2_kernel<<<NB, 256, 0, stream>>>(q_values, fs, g, cb, sb, out);
}